// SelfEquiConv_7739531067651
// MI455X (gfx1250) — compile-verified
//
#include <hip/hip_runtime.h>
#include <hip/hip_bf16.h>
#include <math.h>

// ---------------------------------------------------------------------------
// SelfEquiConv on gfx1250 (MI455X), fp32 WMMA path.
// Each wave32 owns 32 edges (two 16-row M-tiles). Every einsum is a
// (M x K) @ (K x N) GEMM on V_WMMA_F32_16X16X4_F32 where the A operand
// (per-edge outer products) is built on the fly from LDS-staged features.
// Register blocking: K-outer / N-inner (A reused across all N-tiles) and
// 2 M-tiles (each B fragment from L2 feeds two WMMAs).
// ---------------------------------------------------------------------------

typedef float v2f __attribute__((ext_vector_type(2)));
typedef float v8f __attribute__((ext_vector_type(8)));

#define SDIM 64
#define VDIM 32
#define DIN  160               // S + 3V
#define FS   162               // feature tile row stride (even, conflict-free)
#define SACT_ST 66             // silu(z_s) row stride
#define VACT_ST 98             // gated z_v row stride
#define SIGG_ST 34             // sigmoid(z_g) row stride

// scale constants from the reference
#define A_SS_S 1.1048543456039806e-2f   // (2*S*S)^-0.5
#define A_VV_S 1.2757750346101634e-2f   // (2*V*V)^-0.5 / sqrt(3)
#define A_SS_G 1.1048543456039806e-2f
#define A_VV_G 1.2757750346101634e-2f
#define A_SV_V 9.0210896559882004e-3f   // (2*S*V)^-0.5 / sqrt(3)
#define A_VS_V 9.0210896559882004e-3f
#define SCALE_S 0.125f                  // S^-0.5
#define SCALE_V 0.17677669529663687f    // V^-0.5

__device__ __forceinline__ v8f wmma4(v2f a, v2f b, v8f c) {
    return __builtin_amdgcn_wmma_f32_16x16x4_f32(
        false, a, false, b, (short)0, c, false, false);
}

__device__ __forceinline__ float fast_sigmoid(float z) {
    return 1.0f / (1.0f + __expf(-z));
}

__global__ __launch_bounds__(32)
void SelfEquiConv_7739531067651_kernel(
    const float* __restrict__ fea1, const float* __restrict__ fea2,
    const float* __restrict__ w_ss_s, const float* __restrict__ w_vv_s,
    const float* __restrict__ w_ss_g, const float* __restrict__ w_vv_g,
    const float* __restrict__ w_sv_v, const float* __restrict__ w_vs_v,
    const float* __restrict__ wl_s,   const float* __restrict__ wl_v,
    float* __restrict__ out, int E)
{
    // LDS carve-out (floats): f1t | f2t | scratch(sact/vact alias) | sigg
    __shared__ float sm[32*FS + 32*FS + 32*VACT_ST + 32*SIGG_ST];
    float* f1t  = sm;
    float* f2t  = f1t + 32 * FS;
    float* sact = f2t + 32 * FS;       // 32 x 64 (stride 66) -- phase 1
    float* vact = sact;                // 32 x 96 (stride 98) -- phase 2 (alias)
    float* sigg = sact + 32 * VACT_ST; // 32 x 32 (stride 34)

    const int lane = threadIdx.x;            // wave32
    const int e0   = blockIdx.x << 5;        // 32 edges per wave
    const int m    = lane & 15;              // A row within M-tile
    const int kh   = (lane >> 4) << 1;       // K offset: 0 or 2
    const int nl   = lane & 15;              // B/C column in N-tile
    const int mc   = (lane >> 4) << 3;       // C row base: 0 or 8

    // ---- stage 32 edges of both feature tensors into LDS ----
    for (int idx = lane; idx < 32 * DIN; idx += 32) {
        int r = idx / DIN, c = idx - r * DIN;
        int e = e0 + r;
        float a = 0.f, b = 0.f;
        if (e < E) {
            a = fea1[(size_t)e * DIN + c];
            b = fea2[(size_t)e * DIN + c];
        }
        f1t[r * FS + c] = a;
        f2t[r * FS + c] = b;
    }
    __syncthreads();

    const float* r1a = f1t + m * FS;          // M-tile 0 rows
    const float* r2a = f2t + m * FS;
    const float* r1b = f1t + (m + 16) * FS;   // M-tile 1 rows
    const float* r2b = f2t + (m + 16) * FS;

    // ==================  z_s : 32 x 64  (4 N-tiles x 2 M-tiles) ==========
    {
        v8f acc[2][4] = {};
        // ss term: K = 4096, k = u*64+v, ldb = 64
        #pragma unroll 1
        for (int u = 0; u < SDIM; ++u) {
            float s1a = A_SS_S * r1a[u];
            float s1b = A_SS_S * r1b[u];
            const float* brow = w_ss_s + (size_t)(u * SDIM + kh) * SDIM + nl;
            #pragma unroll 2
            for (int v0 = 0; v0 < SDIM; v0 += 4) {
                int v = v0 + kh;
                v2f Aa; Aa.x = s1a * r2a[v]; Aa.y = s1a * r2a[v + 1];
                v2f Ab; Ab.x = s1b * r2b[v]; Ab.y = s1b * r2b[v + 1];
                const float* bp = brow + v0 * SDIM;
                #pragma unroll
                for (int nt = 0; nt < 4; ++nt) {
                    v2f B; B.x = bp[nt * 16]; B.y = bp[SDIM + nt * 16];
                    acc[0][nt] = wmma4(Aa, B, acc[0][nt]);
                    acc[1][nt] = wmma4(Ab, B, acc[1][nt]);
                }
            }
        }
        // vv term: K = 1024, k = u*32+v, A = <v1[u,:],v2[v,:]>
        #pragma unroll 1
        for (int u = 0; u < VDIM; ++u) {
            const float* p1a = r1a + SDIM + u * 3;
            const float* p1b = r1b + SDIM + u * 3;
            const float* brow = w_vv_s + (size_t)(u * VDIM + kh) * SDIM + nl;
            #pragma unroll 2
            for (int v0 = 0; v0 < VDIM; v0 += 4) {
                int v = v0 + kh;
                const float* q2a = r2a + SDIM + v * 3;
                const float* q2b = r2b + SDIM + v * 3;
                v2f Aa, Ab;
                Aa.x = A_VV_S * (p1a[0]*q2a[0] + p1a[1]*q2a[1] + p1a[2]*q2a[2]);
                Aa.y = A_VV_S * (p1a[0]*q2a[3] + p1a[1]*q2a[4] + p1a[2]*q2a[5]);
                Ab.x = A_VV_S * (p1b[0]*q2b[0] + p1b[1]*q2b[1] + p1b[2]*q2b[2]);
                Ab.y = A_VV_S * (p1b[0]*q2b[3] + p1b[1]*q2b[4] + p1b[2]*q2b[5]);
                const float* bp = brow + v0 * SDIM;
                #pragma unroll
                for (int nt = 0; nt < 4; ++nt) {
                    v2f B; B.x = bp[nt * 16]; B.y = bp[SDIM + nt * 16];
                    acc[0][nt] = wmma4(Aa, B, acc[0][nt]);
                    acc[1][nt] = wmma4(Ab, B, acc[1][nt]);
                }
            }
        }
        // SiLU -> sact
        #pragma unroll
        for (int t = 0; t < 2; ++t)
            #pragma unroll
            for (int nt = 0; nt < 4; ++nt)
                #pragma unroll
                for (int r = 0; r < 8; ++r) {
                    float z = acc[t][nt][r];
                    sact[(t*16 + mc + r) * SACT_ST + nt*16 + nl] = z * fast_sigmoid(z);
                }
    }

    // ==================  s_out = silu(z_s) @ wl_s * S^-0.5  ==============
    {
        v8f so[2][4] = {};
        #pragma unroll 2
        for (int k0 = 0; k0 < SDIM; k0 += 4) {
            int k = k0 + kh;
            v2f Aa; Aa.x = sact[m * SACT_ST + k];        Aa.y = sact[m * SACT_ST + k + 1];
            v2f Ab; Ab.x = sact[(m+16) * SACT_ST + k];   Ab.y = sact[(m+16) * SACT_ST + k + 1];
            const float* bp = wl_s + (size_t)k * SDIM + nl - (size_t)(kh ? 0 : 0); // row k
            const float* bq = wl_s + (size_t)(k0 + kh) * SDIM + nl;
            #pragma unroll
            for (int nt = 0; nt < 4; ++nt) {
                v2f B; B.x = bq[nt * 16]; B.y = bq[SDIM + nt * 16];
                so[0][nt] = wmma4(Aa, B, so[0][nt]);
                so[1][nt] = wmma4(Ab, B, so[1][nt]);
            }
            (void)bp;
        }
        #pragma unroll
        for (int t = 0; t < 2; ++t)
            #pragma unroll
            for (int nt = 0; nt < 4; ++nt)
                #pragma unroll
                for (int r = 0; r < 8; ++r) {
                    int e = e0 + t*16 + mc + r;
                    if (e < E) out[(size_t)e * DIN + nt*16 + nl] = SCALE_S * so[t][nt][r];
                }
    }
    __syncthreads();   // sact region will be reused as vact below

    // ==================  z_g : 32 x 32  ==================================
    {
        v8f acc[2][2] = {};
        #pragma unroll 1
        for (int u = 0; u < SDIM; ++u) {
            float s1a = A_SS_G * r1a[u];
            float s1b = A_SS_G * r1b[u];
            const float* brow = w_ss_g + (size_t)(u * SDIM + kh) * VDIM + nl;
            #pragma unroll 2
            for (int v0 = 0; v0 < SDIM; v0 += 4) {
                int v = v0 + kh;
                v2f Aa; Aa.x = s1a * r2a[v]; Aa.y = s1a * r2a[v + 1];
                v2f Ab; Ab.x = s1b * r2b[v]; Ab.y = s1b * r2b[v + 1];
                const float* bp = brow + v0 * VDIM;
                #pragma unroll
                for (int nt = 0; nt < 2; ++nt) {
                    v2f B; B.x = bp[nt * 16]; B.y = bp[VDIM + nt * 16];
                    acc[0][nt] = wmma4(Aa, B, acc[0][nt]);
                    acc[1][nt] = wmma4(Ab, B, acc[1][nt]);
                }
            }
        }
        #pragma unroll 1
        for (int u = 0; u < VDIM; ++u) {
            const float* p1a = r1a + SDIM + u * 3;
            const float* p1b = r1b + SDIM + u * 3;
            const float* brow = w_vv_g + (size_t)(u * VDIM + kh) * VDIM + nl;
            #pragma unroll 2
            for (int v0 = 0; v0 < VDIM; v0 += 4) {
                int v = v0 + kh;
                const float* q2a = r2a + SDIM + v * 3;
                const float* q2b = r2b + SDIM + v * 3;
                v2f Aa, Ab;
                Aa.x = A_VV_G * (p1a[0]*q2a[0] + p1a[1]*q2a[1] + p1a[2]*q2a[2]);
                Aa.y = A_VV_G * (p1a[0]*q2a[3] + p1a[1]*q2a[4] + p1a[2]*q2a[5]);
                Ab.x = A_VV_G * (p1b[0]*q2b[0] + p1b[1]*q2b[1] + p1b[2]*q2b[2]);
                Ab.y = A_VV_G * (p1b[0]*q2b[3] + p1b[1]*q2b[4] + p1b[2]*q2b[5]);
                const float* bp = brow + v0 * VDIM;
                #pragma unroll
                for (int nt = 0; nt < 2; ++nt) {
                    v2f B; B.x = bp[nt * 16]; B.y = bp[VDIM + nt * 16];
                    acc[0][nt] = wmma4(Aa, B, acc[0][nt]);
                    acc[1][nt] = wmma4(Ab, B, acc[1][nt]);
                }
            }
        }
        #pragma unroll
        for (int t = 0; t < 2; ++t)
            #pragma unroll
            for (int nt = 0; nt < 2; ++nt)
                #pragma unroll
                for (int r = 0; r < 8; ++r)
                    sigg[(t*16 + mc + r) * SIGG_ST + nt*16 + nl] =
                        fast_sigmoid(acc[t][nt][r]);
    }

    // ==========  z_v : 32 x 32 x 3, gated by sigmoid(z_g)  ===============
    for (int i = 0; i < 3; ++i) {
        v8f acc[2][2] = {};
        // sv term: k = u(64)*32 + v(32); A = s1[u] * v2[v,i]
        #pragma unroll 1
        for (int u = 0; u < SDIM; ++u) {
            float s1a = A_SV_V * r1a[u];
            float s1b = A_SV_V * r1b[u];
            const float* brow = w_sv_v + (size_t)(u * VDIM + kh) * VDIM + nl;
            #pragma unroll 2
            for (int v0 = 0; v0 < VDIM; v0 += 4) {
                int v = v0 + kh;
                const float* q2a = r2a + SDIM + v * 3 + i;
                const float* q2b = r2b + SDIM + v * 3 + i;
                v2f Aa; Aa.x = s1a * q2a[0]; Aa.y = s1a * q2a[3];
                v2f Ab; Ab.x = s1b * q2b[0]; Ab.y = s1b * q2b[3];
                const float* bp = brow + v0 * VDIM;
                #pragma unroll
                for (int nt = 0; nt < 2; ++nt) {
                    v2f B; B.x = bp[nt * 16]; B.y = bp[VDIM + nt * 16];
                    acc[0][nt] = wmma4(Aa, B, acc[0][nt]);
                    acc[1][nt] = wmma4(Ab, B, acc[1][nt]);
                }
            }
        }
        // vs term: k = u(32)*64 + v(64); A = v1[u,i] * s2[v]
        #pragma unroll 1
        for (int u = 0; u < VDIM; ++u) {
            float v1a = A_VS_V * r1a[SDIM + u * 3 + i];
            float v1b = A_VS_V * r1b[SDIM + u * 3 + i];
            const float* brow = w_vs_v + (size_t)(u * SDIM + kh) * VDIM + nl;
            #pragma unroll 2
            for (int v0 = 0; v0 < SDIM; v0 += 4) {
                int v = v0 + kh;
                v2f Aa; Aa.x = v1a * r2a[v]; Aa.y = v1a * r2a[v + 1];
                v2f Ab; Ab.x = v1b * r2b[v]; Ab.y = v1b * r2b[v + 1];
                const float* bp = brow + v0 * VDIM;
                #pragma unroll
                for (int nt = 0; nt < 2; ++nt) {
                    v2f B; B.x = bp[nt * 16]; B.y = bp[VDIM + nt * 16];
                    acc[0][nt] = wmma4(Aa, B, acc[0][nt]);
                    acc[1][nt] = wmma4(Ab, B, acc[1][nt]);
                }
            }
        }
        #pragma unroll
        for (int t = 0; t < 2; ++t)
            #pragma unroll
            for (int nt = 0; nt < 2; ++nt)
                #pragma unroll
                for (int r = 0; r < 8; ++r) {
                    float g = sigg[(t*16 + mc + r) * SIGG_ST + nt*16 + nl];
                    vact[(t*16 + mc + r) * VACT_ST + (nt*16 + nl) * 3 + i] =
                        g * acc[t][nt][r];
                }
    }
    __syncthreads();

    // ==================  v_out = v_act @ wl_v * V^-0.5  ==================
    for (int i = 0; i < 3; ++i) {
        v8f vo[2][2] = {};
        #pragma unroll 2
        for (int k0 = 0; k0 < VDIM; k0 += 4) {
            int k = k0 + kh;
            v2f Aa; Aa.x = vact[m * VACT_ST + k * 3 + i];
                    Aa.y = vact[m * VACT_ST + (k + 1) * 3 + i];
            v2f Ab; Ab.x = vact[(m+16) * VACT_ST + k * 3 + i];
                    Ab.y = vact[(m+16) * VACT_ST + (k + 1) * 3 + i];
            const float* bp = wl_v + (size_t)k * VDIM + nl;
            #pragma unroll
            for (int nt = 0; nt < 2; ++nt) {
                v2f B; B.x = bp[nt * 16]; B.y = bp[VDIM + nt * 16];
                vo[0][nt] = wmma4(Aa, B, vo[0][nt]);
                vo[1][nt] = wmma4(Ab, B, vo[1][nt]);
            }
        }
        #pragma unroll
        for (int t = 0; t < 2; ++t)
            #pragma unroll
            for (int nt = 0; nt < 2; ++nt)
                #pragma unroll
                for (int r = 0; r < 8; ++r) {
                    int e = e0 + t*16 + mc + r;
                    if (e < E)
                        out[(size_t)e * DIN + SDIM + (nt*16 + nl) * 3 + i] =
                            SCALE_V * vo[t][nt][r];
                }
    }
}

extern "C" void kernel_launch(void* const* d_in, const int* in_sizes, int n_in,
                              void* d_out, int out_size, void* d_ws, size_t ws_size,
                              hipStream_t stream) {
    const float* fea1   = (const float*)d_in[0];
    const float* fea2   = (const float*)d_in[1];
    // d_in[2] (fea_weight) and d_in[3] (batch_edge) are unused by the reference
    const float* w_ss_s = (const float*)d_in[4];
    const float* w_vv_s = (const float*)d_in[5];
    const float* w_ss_g = (const float*)d_in[6];
    const float* w_vv_g = (const float*)d_in[7];
    const float* w_sv_v = (const float*)d_in[8];
    const float* w_vs_v = (const float*)d_in[9];
    const float* wl_s   = (const float*)d_in[10];
    const float* wl_v   = (const float*)d_in[11];
    float* out = (float*)d_out;

    const int E = in_sizes[0] / DIN;          // 100000
    const int blocks = (E + 31) / 32;         // one wave32 per 32-edge tile

    SelfEquiConv_7739531067651_kernel<<<blocks, 32, 0, stream>>>(
        fea1, fea2, w_ss_s, w_vv_s, w_ss_g, w_vv_g,
        w_sv_v, w_vs_v, wl_s, wl_v, out, E);
}